// DiTMoERouter_8761733284135
// MI455X (gfx1250) — compile-verified
//
#include <hip/hip_runtime.h>

// MoE router (DiT): gate GEMM [16384,4096]x[4096,16] (fp32 WMMA 16x16x4),
// softmax(E=16), top-2 + renorm, aux load-balancing loss.
// HBM-bound: 268MB of x @ 23.3TB/s => ~11.5us floor. fp32 WMMA keeps exactness.

typedef __attribute__((ext_vector_type(2))) float v2f;
typedef __attribute__((ext_vector_type(4))) float v4f;
typedef __attribute__((ext_vector_type(8))) float v8f;

#define BB 4
#define SS 4096
#define HH 4096
#define EE 16
#define NTOK (BB * SS)            // 16384 tokens
#define TILES_PER_BLOCK 4         // 4 tiles of 16 tokens => 64 tokens per block
#define NBLOCKS (NTOK / 16 / TILES_PER_BLOCK)  // 256 blocks, 256 thr = 8 waves each

// Wave pairing: waves (2g, 2g+1) split the K=4096 reduction in half for tile g.
// => 2048 active GEMM waves chip-wide for memory-level parallelism.

__global__ __launch_bounds__(256) void moe_router_kernel(
    const float* __restrict__ x,   // [B,S,H] fp32
    const float* __restrict__ W,   // [E,H]  fp32
    float* __restrict__ out,       // [NTOK*2 probs][NTOK*2 idx][1 aux]
    float* __restrict__ ws)        // [NBLOCKS][32] partials: 16 probsum + 16 cnt
{
    __shared__ float tileD[8][16][17];  // per-wave partial logits (pad 17: no bank conflicts)
    __shared__ float colsum[4][16];
    __shared__ int   hist[16];

    const int lane = threadIdx.x & 31;
    const int wave = threadIdx.x >> 5;
    const int grp  = wave >> 1;                      // tile within block: 0..3
    const int half = wave & 1;                       // which K-half
    const int tile = blockIdx.x * TILES_PER_BLOCK + grp;

    // A-matrix 16x4 f32 layout: lane half h, VGPR j -> K = j + 2h (ISA 7.12.2).
    // We load float4 per lane at K-offset 4h and split into two WMMAs; the same
    // K-permutation is applied to B, so products align.
    const int mrow = lane & 15;
    const int koff = ((lane >> 4) << 2) + half * (HH / 2);

    const float* xp = x + (size_t)(tile * 16 + mrow) * HH + koff;
    const float* wp = W + (size_t)mrow * HH + koff;   // B[k][n] = W[n][k], n = mrow

    v8f c0 = {};
    v8f c1 = {};
#pragma unroll 8
    for (int kb = 0; kb < HH / 2; kb += 8) {
        v4f a = *(const v4f*)(xp + kb);
        v4f b = *(const v4f*)(wp + kb);
        v2f a0 = {a.x, a.y}, a1 = {a.z, a.w};
        v2f b0 = {b.x, b.y}, b1 = {b.z, b.w};
        // D = A(16x4 f32) * B(4x16 f32) + C ; two independent accumulators for ILP
        c0 = __builtin_amdgcn_wmma_f32_16x16x4_f32(false, a0, false, b0,
                                                   (short)0, c0, false, false);
        c1 = __builtin_amdgcn_wmma_f32_16x16x4_f32(false, a1, false, b1,
                                                   (short)0, c1, false, false);
    }

    // C/D layout: lane n=lane&15 holds column (expert) n; VGPR r holds row r
    // (lanes 0-15) or r+8 (lanes 16-31). Transpose through LDS.
    {
        const int n  = lane & 15;
        const int mb = (lane >> 4) * 8;
#pragma unroll
        for (int r = 0; r < 8; ++r)
            tileD[wave][mb + r][n] = c0[r] + c1[r];
    }
    if (threadIdx.x < 16) hist[threadIdx.x] = 0;
    __syncthreads();

    // ---- per-token softmax / top-2: 64 threads own the block's 64 tokens ----
    if (threadIdx.x < 64) {
        const int g = threadIdx.x >> 4;
        const int t = threadIdx.x & 15;
        float p[16];
        float mx = -3.0e38f;
#pragma unroll
        for (int e = 0; e < 16; ++e) {
            float v = tileD[2 * g][t][e] + tileD[2 * g + 1][t][e];  // combine K-halves
            p[e] = v;
            mx = fmaxf(mx, v);
        }
        float sum = 0.f;
#pragma unroll
        for (int e = 0; e < 16; ++e) { p[e] = __expf(p[e] - mx); sum += p[e]; }
        const float inv = 1.f / sum;
#pragma unroll
        for (int e = 0; e < 16; ++e) { p[e] *= inv; tileD[2 * g][t][e] = p[e]; }

        // top-2, first-occurrence on ties (matches jax.lax.top_k)
        int i1 = 0; float p1 = p[0];
#pragma unroll
        for (int e = 1; e < 16; ++e) if (p[e] > p1) { p1 = p[e]; i1 = e; }
        int i2 = -1; float p2 = -1.f;
#pragma unroll
        for (int e = 0; e < 16; ++e)
            if (e != i1 && p[e] > p2) { p2 = p[e]; i2 = e; }

        const float z = 1.f / (p1 + p2);          // renormalize top-k probs
        const size_t tok = (size_t)blockIdx.x * 64 + threadIdx.x;
        out[tok * 2 + 0] = p1 * z;
        out[tok * 2 + 1] = p2 * z;
        out[(size_t)NTOK * 2 + tok * 2 + 0] = (float)i1;   // indices as float output
        out[(size_t)NTOK * 2 + tok * 2 + 1] = (float)i2;

        atomicAdd(&hist[i1], 1);   // integer LDS atomics: deterministic
        atomicAdd(&hist[i2], 1);
    }
    __syncthreads();

    // ---- per-expert prob sums over the block's 64 tokens (deterministic) ----
    if (threadIdx.x < 64) {
        const int g = threadIdx.x >> 4;
        const int e = threadIdx.x & 15;
        float s = 0.f;
#pragma unroll
        for (int t = 0; t < 16; ++t) s += tileD[2 * g][t][e];
        colsum[g][e] = s;
    }
    __syncthreads();

    if (threadIdx.x < 16) {
        const int e = threadIdx.x;
        const float s = colsum[0][e] + colsum[1][e] + colsum[2][e] + colsum[3][e];
        ws[(size_t)blockIdx.x * 32 + e]      = s;              // sum of probs
        ws[(size_t)blockIdx.x * 32 + 16 + e] = (float)hist[e]; // selection count
    }
}

__global__ __launch_bounds__(32) void moe_router_finalize(
    const float* __restrict__ ws, float* __restrict__ out)
{
    __shared__ float tot[32];
    const int i = threadIdx.x;
    float s = 0.f;
    for (int b = 0; b < NBLOCKS; ++b) s += ws[(size_t)b * 32 + i];
    tot[i] = s;
    __syncthreads();
    if (i == 0) {
        // aux = E * sum_e (cnt_e / B) * (sum_prob_e / (B*S))
        float aux = 0.f;
        for (int e = 0; e < 16; ++e) {
            const float tokens_pe = tot[16 + e] * (1.0f / (float)BB);
            const float avg_prob  = tot[e] * (1.0f / (float)(BB * SS));
            aux += tokens_pe * avg_prob;
        }
        out[(size_t)NTOK * 2 * 2] = (float)EE * aux;
    }
}

extern "C" void kernel_launch(void* const* d_in, const int* in_sizes, int n_in,
                              void* d_out, int out_size, void* d_ws, size_t ws_size,
                              hipStream_t stream) {
    const float* x = (const float*)d_in[0];   // [4,4096,4096] fp32
    const float* W = (const float*)d_in[1];   // [16,4096] fp32
    float* out = (float*)d_out;               // 65537 floats
    float* ws  = (float*)d_ws;                // needs NBLOCKS*32*4 = 32 KB

    moe_router_kernel<<<NBLOCKS, 256, 0, stream>>>(x, W, out, ws);
    moe_router_finalize<<<1, 32, 0, stream>>>(ws, out);
}